// Model_35828617183861
// MI455X (gfx1250) — compile-verified
//
#include <hip/hip_runtime.h>
#include <hip/hip_bf16.h>
#include <math.h>

typedef __attribute__((ext_vector_type(2))) float v2f;
typedef __attribute__((ext_vector_type(8))) float v8f;

// ===================== utility =====================
__global__ void kzero4(float4* __restrict__ p, long n4) {
  long i = (long)blockIdx.x * blockDim.x + threadIdx.x;
  if (i < n4) p[i] = make_float4(0.f, 0.f, 0.f, 0.f);
}

// Build symmetric binary adjacency, zero diagonal (diag never written).
__global__ void kscatter(float* __restrict__ A, const int* __restrict__ src,
                         const int* __restrict__ dst, int n, long E) {
  long e = (long)blockIdx.x * blockDim.x + threadIdx.x;
  if (e >= E) return;
  int b = (int)(e >> 14);                // 16384 edges per graph
  int s = src[e], d = dst[e];
  if (s == d) return;
  float* Ab = A + (long)b * n * n;
  Ab[(long)s * n + d] = 1.0f;
  Ab[(long)d * n + s] = 1.0f;
}

// Per-row: rowsum s, diag dg.  dis_gcn = rsqrt(s + sl), sl = (dg==0),
// dis0 = (s-dg)>0 ? rsqrt(s-dg) : 0.   One wave per row, 8 waves/block.
__global__ __launch_bounds__(256)
void krowstats(const float* __restrict__ A, int n, float* __restrict__ disg,
               float* __restrict__ slv, float* __restrict__ dis0,
               float* __restrict__ dgl) {
  int gw   = (int)((blockIdx.x * (long)blockDim.x + threadIdx.x) >> 5);
  int lane = threadIdx.x & 31;
  int b = gw / n, i = gw % n;
  const float* row = A + ((long)b * n + i) * n;
  float s = 0.f;
  for (int j = lane; j < n; j += 32) s += row[j];
  for (int off = 16; off; off >>= 1) s += __shfl_down(s, off, 32);
  if (lane == 0) {
    float dg = row[i];
    float sl = (dg == 0.f) ? 1.f : 0.f;
    disg[gw] = rsqrtf(s + sl);
    slv[gw]  = sl;
    float s0 = s - dg;
    dis0[gw] = (s0 > 0.f) ? rsqrtf(s0) : 0.f;
    dgl[gw]  = dg;
  }
}

// ===================== WMMA f32 GEMM =====================
// C[b] (MxN) = A[b] (MxK) * B[b] (KxN); M%128==0, N%64==0, K%32==0.
// Block 256 thr = 8 waves (4x2), wave = 32x32 (2x2 wmma 16x16x4 tiles).
// B tile kept transposed in LDS so both fragments are contiguous K-pairs.
__global__ __launch_bounds__(256)
void kgemm(const float* __restrict__ Ag, long sA, int lda,
           const float* __restrict__ Bg, long sB, int ldb,
           float* __restrict__ Cg, long sC, int ldc, int K) {
  __shared__ float As[128][33];    // odd stride: conflict-free
  __shared__ float BsT[64][33];
  int b = blockIdx.z;
  const float* Ab = Ag + (long)b * sA;
  const float* Bb = Bg + (long)b * sB;
  float*       Cb = Cg + (long)b * sC;
  int m0 = blockIdx.x * 128;
  int n0 = blockIdx.y * 64;
  int tid = threadIdx.x;
  int lane = tid & 31, wid = tid >> 5;
  int wm = (wid >> 1) * 32;        // 0,32,64,96
  int wn = (wid & 1) * 32;         // 0,32
  int hl = lane >> 4;              // lane half: K sub-pair select
  int l15 = lane & 15;
  int ra0 = wm + l15, ra1 = wm + 16 + l15;
  int cb0 = wn + l15, cb1 = wn + 16 + l15;
  v8f acc[2][2] = {};
  for (int k0 = 0; k0 < K; k0 += 32) {
    // ---- stage A tile 128x32 (4 x float4 per thread) ----
#pragma unroll
    for (int i = 0; i < 4; ++i) {
      int q = tid + i * 256;
      int r = q >> 3, c = (q & 7) * 4;
      float4 v = *(const float4*)(Ab + (long)(m0 + r) * lda + (k0 + c));
      As[r][c] = v.x; As[r][c + 1] = v.y; As[r][c + 2] = v.z; As[r][c + 3] = v.w;
    }
    // ---- stage B tile 32x64, transposed into BsT (2 x float4 per thread) ----
#pragma unroll
    for (int i = 0; i < 2; ++i) {
      int q = tid + i * 256;
      int r = q >> 4, c = (q & 15) * 4;
      float4 v = *(const float4*)(Bb + (long)(k0 + r) * ldb + (n0 + c));
      BsT[c][r] = v.x; BsT[c + 1][r] = v.y; BsT[c + 2][r] = v.z; BsT[c + 3][r] = v.w;
    }
    // prefetch next K-chunk while this one is consumed
    if (k0 + 32 < K) {
      __builtin_prefetch(Ab + (long)(m0 + (tid >> 1)) * lda + (k0 + 32) + (tid & 1) * 16, 0, 1);
      __builtin_prefetch(Bb + (long)(k0 + 32 + (tid >> 3)) * ldb + n0 + (tid & 7) * 8, 0, 1);
    }
    __syncthreads();
    // ---- software-pipelined fragment fetch + 2x2 WMMA per 4-wide K step ----
    v2f afn[2], bfn[2];
    afn[0].x = As[ra0][2 * hl];      afn[0].y = As[ra0][2 * hl + 1];
    afn[1].x = As[ra1][2 * hl];      afn[1].y = As[ra1][2 * hl + 1];
    bfn[0].x = BsT[cb0][2 * hl];     bfn[0].y = BsT[cb0][2 * hl + 1];
    bfn[1].x = BsT[cb1][2 * hl];     bfn[1].y = BsT[cb1][2 * hl + 1];
#pragma unroll
    for (int kk = 0; kk < 32; kk += 4) {
      v2f af0 = afn[0], af1 = afn[1], bf0 = bfn[0], bf1 = bfn[1];
      if (kk + 4 < 32) {
        int kb = kk + 4 + 2 * hl;
        afn[0].x = As[ra0][kb];      afn[0].y = As[ra0][kb + 1];
        afn[1].x = As[ra1][kb];      afn[1].y = As[ra1][kb + 1];
        bfn[0].x = BsT[cb0][kb];     bfn[0].y = BsT[cb0][kb + 1];
        bfn[1].x = BsT[cb1][kb];     bfn[1].y = BsT[cb1][kb + 1];
      }
      acc[0][0] = __builtin_amdgcn_wmma_f32_16x16x4_f32(false, af0, false, bf0, (short)0, acc[0][0], false, false);
      acc[0][1] = __builtin_amdgcn_wmma_f32_16x16x4_f32(false, af0, false, bf1, (short)0, acc[0][1], false, false);
      acc[1][0] = __builtin_amdgcn_wmma_f32_16x16x4_f32(false, af1, false, bf0, (short)0, acc[1][0], false, false);
      acc[1][1] = __builtin_amdgcn_wmma_f32_16x16x4_f32(false, af1, false, bf1, (short)0, acc[1][1], false, false);
    }
    __syncthreads();
  }
#pragma unroll
  for (int mi = 0; mi < 2; ++mi)
#pragma unroll
    for (int ni = 0; ni < 2; ++ni)
#pragma unroll
      for (int r = 0; r < 8; ++r) {  // C layout: vgpr r = row r (lanes0-15) / 8+r (lanes16-31)
        int row = m0 + wm + mi * 16 + hl * 8 + r;
        int col = n0 + wn + ni * 16 + l15;
        Cb[(long)row * ldc + col] = acc[mi][ni][r];
      }
}

// ===================== elementwise =====================
__global__ void kscale(float* __restrict__ Y, const float* __restrict__ X,
                       const float* __restrict__ s, long total) {
  long i = (long)blockIdx.x * blockDim.x + threadIdx.x;
  if (i < total) Y[i] = X[i] * s[i >> 7];
}

// out = relu( dis * (T + sl*Hs) + bias )
__global__ void kgcn_epi(float* __restrict__ out, const float* __restrict__ T,
                         const float* __restrict__ Hs, const float* __restrict__ dis,
                         const float* __restrict__ sl, const float* __restrict__ bias,
                         long total) {
  long i = (long)blockIdx.x * blockDim.x + threadIdx.x;
  if (i >= total) return;
  long row = i >> 7; int f = (int)(i & 127);
  float v = dis[row] * (T[i] + sl[row] * Hs[i]) + bias[f];
  out[i] = v > 0.f ? v : 0.f;
}

// score_i = sum_f | X_if - dis0_i * (T_if - dgl_i * Ys_if) | ; wave per node
__global__ __launch_bounds__(256)
void kinfo(float* __restrict__ score, const float* __restrict__ X,
           const float* __restrict__ T, const float* __restrict__ Ys,
           const float* __restrict__ dis0, const float* __restrict__ dgl) {
  int node = (int)((blockIdx.x * (long)blockDim.x + threadIdx.x) >> 5);
  int lane = threadIdx.x & 31;
  long base = (long)node << 7;
  float d0 = dis0[node], dg = dgl[node];
  float s = 0.f;
  for (int f = lane; f < 128; f += 32) {
    float t = T[base + f] - dg * Ys[base + f];
    s += fabsf(X[base + f] - d0 * t);
  }
  for (int off = 16; off; off >>= 1) s += __shfl_down(s, off, 32);
  if (lane == 0) score[node] = s;
}

// ===================== top-k (bitonic, descending, stable ties) =====================
__global__ __launch_bounds__(1024)
void ktopk(int* __restrict__ perm, const float* __restrict__ score, int n, int k) {
  __shared__ float sv[1024];
  __shared__ int   si[1024];
  int b = blockIdx.x, t = threadIdx.x;
  if (t < n) { sv[t] = score[(long)b * n + t]; si[t] = t; }
  __syncthreads();
  for (int len = 2; len <= n; len <<= 1)
    for (int stride = len >> 1; stride > 0; stride >>= 1) {
      int i = t, j = i ^ stride;
      if (j > i && j < n) {
        float vi = sv[i], vj = sv[j];
        int ii = si[i], ij = si[j];
        bool iFirst = (vi > vj) || (vi == vj && ii < ij);  // "comes first" in descending
        bool desc = ((i & len) == 0);
        if (desc ? !iFirst : iFirst) { sv[i] = vj; sv[j] = vi; si[i] = ij; si[j] = ii; }
      }
      __syncthreads();
    }
  if (t < k) perm[(long)b * k + t] = si[t];
}

__global__ void kgather(float* __restrict__ Xp, const float* __restrict__ X,
                        const int* __restrict__ perm, int n, int k, long total) {
  long i = (long)blockIdx.x * blockDim.x + threadIdx.x;
  if (i >= total) return;
  long row = i >> 7; int f = (int)(i & 127);
  int b = (int)(row / k), t = (int)(row % k);
  Xp[i] = X[((long)b * n + perm[(long)b * k + t]) * 128 + f];
}

// a_row / a_col attention dots; wave per node
__global__ __launch_bounds__(256)
void katt(float* __restrict__ arow, float* __restrict__ acol,
          const float* __restrict__ Xp, const float* __restrict__ att) {
  int node = (int)((blockIdx.x * (long)blockDim.x + threadIdx.x) >> 5);
  int lane = threadIdx.x & 31;
  long base = (long)node << 7;
  float s1 = 0.f, s2 = 0.f;
  for (int f = lane; f < 128; f += 32) {
    float v = Xp[base + f];
    s1 += v * att[f];
    s2 += v * att[128 + f];
  }
  for (int off = 16; off; off >>= 1) {
    s1 += __shfl_down(s1, off, 32);
    s2 += __shfl_down(s2, off, 32);
  }
  if (lane == 0) { arow[node] = s1; acol[node] = s2; }
}

__device__ __forceinline__ float blkMax256(float v, float* red) {
  for (int off = 16; off; off >>= 1) v = fmaxf(v, __shfl_down(v, off, 32));
  int lane = threadIdx.x & 31, wid = threadIdx.x >> 5;
  __syncthreads();
  if (lane == 0) red[wid] = v;
  __syncthreads();
  float r = red[0];
#pragma unroll
  for (int w = 1; w < 8; ++w) r = fmaxf(r, red[w]);
  return r;
}
__device__ __forceinline__ float blkSum256(float v, float* red) {
  for (int off = 16; off; off >>= 1) v += __shfl_down(v, off, 32);
  int lane = threadIdx.x & 31, wid = threadIdx.x >> 5;
  __syncthreads();
  if (lane == 0) red[wid] = v;
  __syncthreads();
  float r = red[0];
#pragma unroll
  for (int w = 1; w < 8; ++w) r += red[w];
  return r;
}

// Row of structure-learned adjacency: softmax_j( leakyrelu(ar_i+ac_j,0.2) + 1.0*Aold[pi,pj] )
__global__ __launch_bounds__(256)
void kpoolrow(float* __restrict__ Anew, const float* __restrict__ arow,
              const float* __restrict__ acol, const float* __restrict__ Aold,
              const int* __restrict__ perm, int nold, int k) {
  __shared__ float red[8];
  int row = blockIdx.x;            // b*k + i
  int b = row / k;
  int t = threadIdx.x;
  const int* pb = perm + (long)b * k;
  int pi = pb[row % k];
  const float* Ar = Aold + ((long)b * nold + pi) * nold;
  float ar = arow[row];
  float zbuf[2];
  int cnt = 0;
  float lmax = -INFINITY;
  for (int j = t; j < k; j += 256) {
    float z = ar + acol[(long)b * k + j];
    z = z > 0.f ? z : 0.2f * z;
    z += Ar[pb[j]];                // LAMB = 1.0
    zbuf[cnt++] = z;
    lmax = fmaxf(lmax, z);
  }
  float m = blkMax256(lmax, red);
  float lsum = 0.f;
  for (int c = 0; c < cnt; ++c) { zbuf[c] = expf(zbuf[c] - m); lsum += zbuf[c]; }
  float s = blkSum256(lsum, red);
  float inv = 1.f / s;
  cnt = 0;
  for (int j = t; j < k; j += 256) Anew[(long)row * k + j] = zbuf[cnt++] * inv;
}

// readout: [max_t | mean_t] over nodes; block 128, grid B
__global__ __launch_bounds__(128)
void kreadout(float* __restrict__ xr, const float* __restrict__ X, int n) {
  int b = blockIdx.x, f = threadIdx.x;
  const float* Xb = X + (long)b * n * 128;
  float mx = -INFINITY, sm = 0.f;
  for (int t = 0; t < n; ++t) {
    float v = Xb[(long)t * 128 + f];
    mx = fmaxf(mx, v); sm += v;
  }
  xr[b * 256 + f] = mx;
  xr[b * 256 + 128 + f] = sm / (float)n;
}

__global__ void klinear(float* __restrict__ out, const float* __restrict__ in,
                        const float* __restrict__ W, const float* __restrict__ bias,
                        int K, int O, int relu) {
  int b = blockIdx.x;
  for (int o = threadIdx.x; o < O; o += blockDim.x) {
    float s = bias[o];
    for (int kk = 0; kk < K; ++kk) s += in[b * K + kk] * W[kk * O + o];
    out[b * O + o] = (relu && s < 0.f) ? 0.f : s;
  }
}

__global__ __launch_bounds__(384)
void kcombine(float* __restrict__ g, const float* __restrict__ x1,
              const float* __restrict__ x2, const float* __restrict__ x3,
              const float* __restrict__ xsk) {
  int b = blockIdx.x, t = threadIdx.x;
  float v;
  if (t < 256)
    v = fmaxf(x1[b * 256 + t], 0.f) + fmaxf(x2[b * 256 + t], 0.f) + fmaxf(x3[b * 256 + t], 0.f);
  else
    v = xsk[b * 128 + (t - 256)];
  g[b * 384 + t] = v;
}

__global__ void klogsm(float* __restrict__ out, const float* __restrict__ z) {
  int b = blockIdx.x * blockDim.x + threadIdx.x;
  if (b >= 64) return;
  const float* r = z + b * 10;
  float m = r[0];
  for (int i = 1; i < 10; ++i) m = fmaxf(m, r[i]);
  float s = 0.f;
  for (int i = 0; i < 10; ++i) s += expf(r[i] - m);
  float l = logf(s);
  for (int i = 0; i < 10; ++i) out[b * 10 + i] = r[i] - m - l;
}

// ===================== host =====================
static void gemm(const float* A, long sA, int lda, const float* B, long sB, int ldb,
                 float* C, long sC, int ldc, int M, int N, int K, hipStream_t st) {
  dim3 grid(M / 128, N / 64, 64);
  kgemm<<<grid, 256, 0, st>>>(A, sA, lda, B, sB, ldb, C, sC, ldc, K);
}

extern "C" void kernel_launch(void* const* d_in, const int* in_sizes, int n_in,
                              void* d_out, int out_size, void* d_ws, size_t ws_size,
                              hipStream_t stream) {
  (void)in_sizes; (void)n_in; (void)out_size; (void)ws_size;
  const float* x    = (const float*)d_in[0];
  const int*   src  = (const int*)d_in[1];
  const int*   dst  = (const int*)d_in[2];
  const float* skew = (const float*)d_in[3];
  const float* W1 = (const float*)d_in[4],  *b1 = (const float*)d_in[5];
  const float* W2 = (const float*)d_in[6],  *b2 = (const float*)d_in[7];
  const float* W3 = (const float*)d_in[8],  *b3 = (const float*)d_in[9];
  const float* att1 = (const float*)d_in[10], *att2 = (const float*)d_in[11];
  const float* Wsk = (const float*)d_in[12], *bsk = (const float*)d_in[13];
  const float* Wl1 = (const float*)d_in[14], *bl1 = (const float*)d_in[15];
  const float* Wl2 = (const float*)d_in[16], *bl2 = (const float*)d_in[17];
  const float* Wl3 = (const float*)d_in[18], *bl3 = (const float*)d_in[19];
  float* out = (float*)d_out;

  const int Bn = 64, N1 = 1024, K1 = 512, K2 = 256;
  size_t off = 0;
  float* ws = (float*)d_ws;
  auto take = [&](size_t n) { float* p = ws + off; off += n; return p; };
  float* A    = take((size_t)Bn * N1 * N1);       // 256 MB
  float* A1   = take((size_t)Bn * K1 * K1);       // 64 MB
  float* A2   = take((size_t)Bn * K2 * K2);       // 16 MB
  float* H    = take((size_t)Bn * N1 * 128);
  float* Hs   = take((size_t)Bn * N1 * 128);
  float* T    = take((size_t)Bn * N1 * 128);
  float* Xb   = take((size_t)Bn * N1 * 128);      // X1 -> X2 -> X3
  float* Xp   = take((size_t)Bn * K1 * 128);      // Xp1 -> Xp2
  float* disg = take((size_t)Bn * N1);
  float* slv  = take((size_t)Bn * N1);
  float* dis0 = take((size_t)Bn * N1);
  float* dgl  = take((size_t)Bn * N1);
  float* scor = take((size_t)Bn * N1);
  int*   perm = (int*)take((size_t)Bn * N1);
  float* arow = take((size_t)Bn * N1);
  float* acol = take((size_t)Bn * N1);
  float* x1r  = take((size_t)Bn * 256);
  float* x2r  = take((size_t)Bn * 256);
  float* x3r  = take((size_t)Bn * 256);
  float* xsk  = take((size_t)Bn * 128);
  float* g    = take((size_t)Bn * 384);
  float* g1   = take((size_t)Bn * 128);
  float* g2   = take((size_t)Bn * 64);
  float* logi = take((size_t)Bn * 10);

  // ---- build adjacency ----
  long nA4 = (long)Bn * N1 * N1 / 4;
  kzero4<<<(unsigned)(nA4 / 256), 256, 0, stream>>>((float4*)A, nA4);
  long E = (long)Bn * 16384;
  kscatter<<<(unsigned)(E / 256), 256, 0, stream>>>(A, src, dst, N1, E);

  // ================= Level 1 (n=1024) =================
  long t1 = (long)Bn * N1 * 128;
  krowstats<<<Bn * N1 / 8, 256, 0, stream>>>(A, N1, disg, slv, dis0, dgl);
  gemm(x, (long)N1 * 128, 128, W1, 0, 128, H, (long)N1 * 128, 128, N1, 128, 128, stream);
  kscale<<<(unsigned)(t1 / 256), 256, 0, stream>>>(Hs, H, disg, t1);
  gemm(A, (long)N1 * N1, N1, Hs, (long)N1 * 128, 128, T, (long)N1 * 128, 128, N1, 128, N1, stream);
  kgcn_epi<<<(unsigned)(t1 / 256), 256, 0, stream>>>(Xb, T, Hs, disg, slv, b1, t1);
  // pool 1
  kscale<<<(unsigned)(t1 / 256), 256, 0, stream>>>(Hs, Xb, dis0, t1);   // Ys
  gemm(A, (long)N1 * N1, N1, Hs, (long)N1 * 128, 128, T, (long)N1 * 128, 128, N1, 128, N1, stream);
  kinfo<<<Bn * N1 / 8, 256, 0, stream>>>(scor, Xb, T, Hs, dis0, dgl);
  ktopk<<<Bn, N1, 0, stream>>>(perm, scor, N1, K1);
  long tp1 = (long)Bn * K1 * 128;
  kgather<<<(unsigned)(tp1 / 256), 256, 0, stream>>>(Xp, Xb, perm, N1, K1, tp1);
  kreadout<<<Bn, 128, 0, stream>>>(x1r, Xp, K1);
  katt<<<Bn * K1 / 8, 256, 0, stream>>>(arow, acol, Xp, att1);
  kpoolrow<<<Bn * K1, 256, 0, stream>>>(A1, arow, acol, A, perm, N1, K1);

  // ================= Level 2 (n=512) =================
  long t2 = (long)Bn * K1 * 128;
  krowstats<<<Bn * K1 / 8, 256, 0, stream>>>(A1, K1, disg, slv, dis0, dgl);
  gemm(Xp, (long)K1 * 128, 128, W2, 0, 128, H, (long)K1 * 128, 128, K1, 128, 128, stream);
  kscale<<<(unsigned)(t2 / 256), 256, 0, stream>>>(Hs, H, disg, t2);
  gemm(A1, (long)K1 * K1, K1, Hs, (long)K1 * 128, 128, T, (long)K1 * 128, 128, K1, 128, K1, stream);
  kgcn_epi<<<(unsigned)(t2 / 256), 256, 0, stream>>>(Xb, T, Hs, disg, slv, b2, t2);  // X2
  // pool 2
  kscale<<<(unsigned)(t2 / 256), 256, 0, stream>>>(Hs, Xb, dis0, t2);
  gemm(A1, (long)K1 * K1, K1, Hs, (long)K1 * 128, 128, T, (long)K1 * 128, 128, K1, 128, K1, stream);
  kinfo<<<Bn * K1 / 8, 256, 0, stream>>>(scor, Xb, T, Hs, dis0, dgl);
  ktopk<<<Bn, K1, 0, stream>>>(perm, scor, K1, K2);
  long tp2 = (long)Bn * K2 * 128;
  kgather<<<(unsigned)(tp2 / 256), 256, 0, stream>>>(Xp, Xb, perm, K1, K2, tp2);    // Xp2
  kreadout<<<Bn, 128, 0, stream>>>(x2r, Xp, K2);
  katt<<<Bn * K2 / 8, 256, 0, stream>>>(arow, acol, Xp, att2);
  kpoolrow<<<Bn * K2, 256, 0, stream>>>(A2, arow, acol, A1, perm, K1, K2);

  // ================= Level 3 (n=256) =================
  long t3 = (long)Bn * K2 * 128;
  krowstats<<<Bn * K2 / 8, 256, 0, stream>>>(A2, K2, disg, slv, dis0, dgl);
  gemm(Xp, (long)K2 * 128, 128, W3, 0, 128, H, (long)K2 * 128, 128, K2, 128, 128, stream);
  kscale<<<(unsigned)(t3 / 256), 256, 0, stream>>>(Hs, H, disg, t3);
  gemm(A2, (long)K2 * K2, K2, Hs, (long)K2 * 128, 128, T, (long)K2 * 128, 128, K2, 128, K2, stream);
  kgcn_epi<<<(unsigned)(t3 / 256), 256, 0, stream>>>(Xb, T, Hs, disg, slv, b3, t3);  // X3
  kreadout<<<Bn, 128, 0, stream>>>(x3r, Xb, K2);

  // ================= head =================
  klinear<<<Bn, 128, 0, stream>>>(xsk, skew, Wsk, bsk, 64, 128, 1);
  kcombine<<<Bn, 384, 0, stream>>>(g, x1r, x2r, x3r, xsk);
  klinear<<<Bn, 128, 0, stream>>>(g1, g, Wl1, bl1, 384, 128, 1);
  klinear<<<Bn, 128, 0, stream>>>(g2, g1, Wl2, bl2, 128, 64, 1);
  klinear<<<Bn, 128, 0, stream>>>(logi, g2, Wl3, bl3, 64, 10, 0);
  klogsm<<<1, 64, 0, stream>>>(out, logi);
}